// Spectral_Discriminator_11364483465245
// MI455X (gfx1250) — compile-verified
//
#include <hip/hip_runtime.h>
#include <math.h>

// ---------------------------------------------------------------------------
// Spectral discriminator on MI455X (gfx1250, wave32).
// DFT-as-GEMM with V_WMMA_F32_16X16X4_F32 (f32 matrix pipe).
// Stage 2 stages its B panels into LDS with the Tensor Data Mover
// (tensor_load_to_lds, TENSORcnt), double-buffered, and feeds WMMA from LDS.
// Fused log-magnitude + radial binning epilogue (ds_add_f32).
// ---------------------------------------------------------------------------

#define HH        256
#define NPIX      (HH * HH)          // 65536
#define BATCH     128
#define NBINS     181                // floor(hypot(127.5,127.5)) + 1
#define NBINS_PAD 192
#define THRESH    90                 // int(256 / (2*sqrt(2)))
#define EPSF      1e-8f
#define KC        32                 // K-chunk staged per TDM transfer

typedef __attribute__((ext_vector_type(2))) float v2f;
typedef __attribute__((ext_vector_type(8))) float v8f;
typedef __attribute__((ext_vector_type(4))) unsigned int v4u;
typedef __attribute__((ext_vector_type(8))) unsigned int v8u;

// Workspace layout (units: floats)
constexpr size_t OFF_C      = 0;                          // 256x256 cos table
constexpr size_t OFF_S      = OFF_C + NPIX;               // 256x256 sin table
constexpr size_t OFF_BIN    = OFF_S + NPIX;               // 256x256 int bin map
constexpr size_t OFF_COUNTS = OFF_BIN + NPIX;             // 256 (pad of 181)
constexpr size_t OFF_SUMS   = OFF_COUNTS + 256;           // 128 x 192
constexpr size_t OFF_GRAY   = OFF_SUMS + (size_t)BATCH * NBINS_PAD;
constexpr size_t OFF_TRE    = OFF_GRAY + (size_t)BATCH * NPIX;
constexpr size_t OFF_TIM    = OFF_TRE + (size_t)BATCH * NPIX;

// ---------------------------------------------------------------------------
__global__ __launch_bounds__(256) void zero_kernel(float* p, int n) {
    int i = blockIdx.x * 256 + threadIdx.x;
    if (i < n) p[i] = 0.0f;
}

// Build DFT tables (exact phase via jk mod 256), fftshifted radial bin map,
// and per-bin pixel counts.
__global__ __launch_bounds__(256) void init_tables_kernel(float* Cm, float* Sm,
                                                          int* bin_map, float* counts) {
    int tid = blockIdx.x * 256 + threadIdx.x;   // 0 .. 65535
    int j = tid >> 8;
    int k = tid & 255;
    int phase = (j * k) & 255;                  // exp(-2*pi*i*jk/256) periodic mod 256
    float theta = (2.0f * 3.14159265358979323846f / 256.0f) * (float)phase;
    Cm[tid] = cosf(theta);
    Sm[tid] = sinf(theta);

    // fftshift: output index i <- input index (i + 128) mod 256
    int ys = (j + 128) & 255;
    int xs = (k + 128) & 255;
    float dy = (float)ys - 127.5f;
    float dx = (float)xs - 127.5f;
    int bin = (int)sqrtf(dy * dy + dx * dx);
    bin_map[tid] = bin;
    atomicAdd(&counts[bin], 1.0f);
}

// gray = 0.2989 R + 0.587 G + 0.114 B
__global__ __launch_bounds__(256) void gray_kernel(const float* __restrict__ x,
                                                   float* __restrict__ gray) {
    size_t idx = (size_t)blockIdx.x * 256 + threadIdx.x;   // over BATCH*NPIX
    size_t b = idx / NPIX;
    size_t p = idx - b * NPIX;
    const float* base = x + (b * 3) * (size_t)NPIX + p;
    gray[idx] = 0.2989f * base[0] + 0.587f * base[NPIX] + 0.114f * base[2 * NPIX];
}

// ---------------------------------------------------------------------------
// Tensor Data Mover: DMA a KC x 128 f32 tile (row stride 256 elements in
// global) to LDS. Descriptor per CDNA5 ISA ch.8 (2 groups, VADDR2/3 NULL):
//   group0: [1:0]=count=1, [63:32]=lds_addr, [120:64]=global_addr, [127:126]=2
//   group1: [17:16]=data_size=2 (4B), [79:48]=tensor_dim0=256,
//           [111:80]=tensor_dim1=256, [127:112]=tile_dim0=128,
//           [143:128]=tile_dim1=KC, [207:160]=tensor_dim0_stride=256
// ---------------------------------------------------------------------------
__device__ __forceinline__ void tdm_load_panel(unsigned long long gaddr,
                                               unsigned int lds_off) {
    v4u g0 = { 1u,
               lds_off,
               (unsigned int)(gaddr & 0xffffffffull),
               (unsigned int)((gaddr >> 32) & 0x1ffffffull) | (2u << 30) };
    v8u g1 = { 0x00020000u,           // data_size = 4 bytes
               (256u << 16),          // tensor_dim0 = 256 (bits 79:48, low half)
               (256u << 16),          // tensor_dim1 = 256 (bits 111:80, low half)
               (128u << 16),          // tile_dim0 = 128   (bits 127:112)
               (unsigned int)KC,      // tile_dim1 = KC    (bits 143:128)
               256u,                  // tensor_dim0_stride = 256 (bits 191:160)
               0u, 0u };
    asm volatile("tensor_load_to_lds %0, %1" :: "s"(g0), "s"(g1) : "memory");
}

// ---------------------------------------------------------------------------
// Stage 1: T = G * (C - iS)   =>  T_re = G*C , T_im = -(G*S)
// One wave computes one 16x16 f32 tile, K stepped by 4 (WMMA f32 16x16x4).
// Operands stream straight from L2 (gray + DFT tables are L2-resident).
// ---------------------------------------------------------------------------
__global__ __launch_bounds__(256) void dft_stage1_kernel(const float* __restrict__ gray,
                                                         const float* __restrict__ Cm,
                                                         const float* __restrict__ Sm,
                                                         float* __restrict__ Tre,
                                                         float* __restrict__ Tim) {
    const int wave = threadIdx.x >> 5;
    const int lane = threadIdx.x & 31;
    const int tile = blockIdx.x * 8 + wave;        // 0..255
    const int m0 = (tile >> 4) << 4;
    const int n0 = (tile & 15) << 4;
    const int b  = blockIdx.y;

    const float* A = gray + (size_t)b * NPIX;
    const int row  = m0 + (lane & 15);
    const int koff = (lane >> 4) * 2;
    const int ncol = n0 + (lane & 15);

    v8f accR = {};
    v8f accI = {};

    for (int k = 0; k < 256; k += 4) {
        __builtin_prefetch(A + row * 256 + k + 32, 0, 1);
        v2f a = *(const v2f*)(A + row * 256 + k + koff);
        v2f bc, bs;
        bc.x = Cm[(k + koff + 0) * 256 + ncol];
        bc.y = Cm[(k + koff + 1) * 256 + ncol];
        bs.x = Sm[(k + koff + 0) * 256 + ncol];
        bs.y = Sm[(k + koff + 1) * 256 + ncol];
        accR = __builtin_amdgcn_wmma_f32_16x16x4_f32(false, a, false, bc,
                                                     (short)0, accR, false, false);
        accI = __builtin_amdgcn_wmma_f32_16x16x4_f32(false, a, false, bs,
                                                     (short)0, accI, false, false);
    }

    float* oR = Tre + (size_t)b * NPIX;
    float* oI = Tim + (size_t)b * NPIX;
    const int rbase = m0 + (lane >> 4) * 8;
    const int c = n0 + (lane & 15);
#pragma unroll
    for (int v = 0; v < 8; ++v) {
        oR[(rbase + v) * 256 + c] = accR[v];
        oI[(rbase + v) * 256 + c] = -accI[v];
    }
}

// ---------------------------------------------------------------------------
// Stage 2: F = (C - iS) * T
//   F_re = C*T_re + S*T_im   (chained into one accumulator)
//   F_im = C*T_im - S*T_re   (two accumulators; f32 WMMA has no A/B negate)
// Each block computes a 16x128 output patch; the shared B panels (T_re, T_im,
// KC x 128) are DMA'd to LDS by the TDM, double-buffered, one wave driving.
// Fused epilogue: mag = log(sqrt(re^2+im^2)+eps), LDS radial binning.
// ---------------------------------------------------------------------------
__global__ __launch_bounds__(256) void dft_stage2_kernel(const float* __restrict__ Cm,
                                                         const float* __restrict__ Sm,
                                                         const float* __restrict__ Tre,
                                                         const float* __restrict__ Tim,
                                                         const int* __restrict__ bin_map,
                                                         float* __restrict__ sums) {
    __shared__ float ldsRe[2][KC * 128];
    __shared__ float ldsIm[2][KC * 128];
    __shared__ float lbins[NBINS_PAD];
    for (int i = threadIdx.x; i < NBINS_PAD; i += 256) lbins[i] = 0.0f;

    const int wave  = threadIdx.x >> 5;
    const int lane  = threadIdx.x & 31;
    const int m0    = (blockIdx.x >> 1) << 4;     // 16 output rows per block
    const int n0blk = (blockIdx.x & 1) << 7;      // 128 output cols per block
    const int b     = blockIdx.y;

    const int row  = m0 + (lane & 15);
    const int koff = (lane >> 4) * 2;
    const int lcol = (wave << 4) + (lane & 15);   // local col 0..127
    const int ncol = n0blk + lcol;

    const unsigned long long gRe =
        (unsigned long long)(Tre + (size_t)b * NPIX + n0blk);
    const unsigned long long gIm =
        (unsigned long long)(Tim + (size_t)b * NPIX + n0blk);
    const unsigned int reOff0 = (unsigned int)(unsigned long long)&ldsRe[0][0];
    const unsigned int reOff1 = (unsigned int)(unsigned long long)&ldsRe[1][0];
    const unsigned int imOff0 = (unsigned int)(unsigned long long)&ldsIm[0][0];
    const unsigned int imOff1 = (unsigned int)(unsigned long long)&ldsIm[1][0];

    v8f accR = {};   // C*T_re + S*T_im
    v8f accA = {};   // C*T_im
    v8f accB = {};   // S*T_re

    // Prime the pipeline: chunk 0 into buffer 0 (wave 0 drives the TDM).
    if (threadIdx.x < 32) {
        tdm_load_panel(gRe, reOff0);
        tdm_load_panel(gIm, imOff0);
    }

    const int nchunks = 256 / KC;                 // 8
    for (int ci = 0; ci < nchunks; ++ci) {
        __syncthreads();                          // buffer (ci+1)&1 is free now
        if (threadIdx.x < 32) {
            if (ci + 1 < nchunks) {
                unsigned long long goff =
                    (unsigned long long)(ci + 1) * KC * 256 * 4;
                tdm_load_panel(gRe + goff, ((ci + 1) & 1) ? reOff1 : reOff0);
                tdm_load_panel(gIm + goff, ((ci + 1) & 1) ? imOff1 : imOff0);
                // 2 newest TDM ops may stay in flight; chunk ci's are done.
                __builtin_amdgcn_s_wait_tensorcnt(2);
            } else {
                __builtin_amdgcn_s_wait_tensorcnt(0);
            }
        }
        __syncthreads();                          // chunk ci landed in LDS

        const float* bufRe = ldsRe[ci & 1];
        const float* bufIm = ldsIm[ci & 1];
        const int kglob = ci * KC;
#pragma unroll
        for (int kk = 0; kk < KC; kk += 4) {
            v2f aC = *(const v2f*)(Cm + row * 256 + kglob + kk + koff);
            v2f aS = *(const v2f*)(Sm + row * 256 + kglob + kk + koff);
            v2f bR, bI;
            bR.x = bufRe[(kk + koff + 0) * 128 + lcol];
            bR.y = bufRe[(kk + koff + 1) * 128 + lcol];
            bI.x = bufIm[(kk + koff + 0) * 128 + lcol];
            bI.y = bufIm[(kk + koff + 1) * 128 + lcol];
            accR = __builtin_amdgcn_wmma_f32_16x16x4_f32(false, aC, false, bR,
                                                         (short)0, accR, false, false);
            accR = __builtin_amdgcn_wmma_f32_16x16x4_f32(false, aS, false, bI,
                                                         (short)0, accR, false, false);
            accA = __builtin_amdgcn_wmma_f32_16x16x4_f32(false, aC, false, bI,
                                                         (short)0, accA, false, false);
            accB = __builtin_amdgcn_wmma_f32_16x16x4_f32(false, aS, false, bR,
                                                         (short)0, accB, false, false);
        }
    }

    const int rbase = m0 + (lane >> 4) * 8;
#pragma unroll
    for (int v = 0; v < 8; ++v) {
        float re = accR[v];
        float im = accA[v] - accB[v];
        float mag = logf(sqrtf(re * re + im * im) + EPSF);
        int bin = bin_map[(rbase + v) * 256 + ncol];
        atomicAdd(&lbins[bin], mag);              // ds_add_f32
    }
    __syncthreads();
    for (int i = threadIdx.x; i < NBINS; i += 256)
        atomicAdd(&sums[(size_t)b * NBINS_PAD + i], lbins[i]);
}

// ---------------------------------------------------------------------------
// Per-image: prof = sums/counts; min/max normalize; NaN->0; dot(last 90, w)+b.
// ---------------------------------------------------------------------------
__global__ __launch_bounds__(256) void finalize_kernel(const float* __restrict__ sums,
                                                       const float* __restrict__ counts,
                                                       const float* __restrict__ w,
                                                       const float* __restrict__ bias,
                                                       float* __restrict__ out) {
    __shared__ float prof[NBINS_PAD];
    __shared__ float red[256];
    const int b = blockIdx.x;
    const int t = threadIdx.x;

    if (t < NBINS) prof[t] = sums[(size_t)b * NBINS_PAD + t] / counts[t];
    __syncthreads();

    float v = (t < NBINS) ? prof[t] : prof[0];

    red[t] = v;
    __syncthreads();
    for (int s = 128; s; s >>= 1) {
        if (t < s) red[t] = fminf(red[t], red[t + s]);
        __syncthreads();
    }
    float mn = red[0];
    __syncthreads();

    red[t] = v;
    __syncthreads();
    for (int s = 128; s; s >>= 1) {
        if (t < s) red[t] = fmaxf(red[t], red[t + s]);
        __syncthreads();
    }
    float mx = red[0];
    __syncthreads();

    float acc = 0.0f;
    if (t < THRESH) {
        float f = (prof[NBINS - THRESH + t] - mn) / (mx - mn);
        if (f != f) f = 0.0f;                 // NaN -> 0
        acc = f * w[t];
    }
    red[t] = acc;
    __syncthreads();
    for (int s = 128; s; s >>= 1) {
        if (t < s) red[t] += red[t + s];
        __syncthreads();
    }
    if (t == 0) out[b] = red[0] + bias[0];
}

// ---------------------------------------------------------------------------
extern "C" void kernel_launch(void* const* d_in, const int* in_sizes, int n_in,
                              void* d_out, int out_size, void* d_ws, size_t ws_size,
                              hipStream_t stream) {
    const float* x    = (const float*)d_in[0];   // (128,3,256,256)
    const float* w    = (const float*)d_in[1];   // (1,90)
    const float* bias = (const float*)d_in[2];   // (1,)
    float* out = (float*)d_out;                  // (128,1)

    float* ws      = (float*)d_ws;
    float* Cm      = ws + OFF_C;
    float* Sm      = ws + OFF_S;
    int*   bin_map = (int*)(ws + OFF_BIN);
    float* counts  = ws + OFF_COUNTS;
    float* sums    = ws + OFF_SUMS;
    float* gray    = ws + OFF_GRAY;
    float* Tre     = ws + OFF_TRE;
    float* Tim     = ws + OFF_TIM;

    // Zero counts + bin sums (contiguous region starting at OFF_COUNTS).
    const int nzero = 256 + BATCH * NBINS_PAD;
    zero_kernel<<<(nzero + 255) / 256, 256, 0, stream>>>(counts, nzero);

    init_tables_kernel<<<NPIX / 256, 256, 0, stream>>>(Cm, Sm, bin_map, counts);

    gray_kernel<<<(BATCH * NPIX) / 256, 256, 0, stream>>>(x, gray);

    dim3 gemm_grid(32, BATCH);   // 256 tiles/image / 8 waves per block
    dft_stage1_kernel<<<gemm_grid, 256, 0, stream>>>(gray, Cm, Sm, Tre, Tim);
    dft_stage2_kernel<<<gemm_grid, 256, 0, stream>>>(Cm, Sm, Tre, Tim, bin_map, sums);

    finalize_kernel<<<BATCH, 256, 0, stream>>>(sums, counts, w, bias, out);
}